// Network_56873956933906
// MI455X (gfx1250) — compile-verified
//
#include <hip/hip_runtime.h>
#include <hip/hip_bf16.h>
#include <math.h>

// ---------------------------------------------------------------------------
// Tensor-field network (e3nn-like), 3 layers, l<=3, fp32 end-to-end.
// MI455X notes: problem is ~4 GFLOP / <3MB live data -> latency bound, not
// roofline bound. We use the fp32 tensor op V_WMMA_F32_16X16X4_F32 for the
// per-atom weight contraction (exact fp32, matrix pipe), keep the S-moments
// resident in LDS (58.4KB < 320KB/WGP), and fuse gating into the layer kernel.
// Round-2 fixes: wave-uniform unit selection via readfirstlane (no EXEC
// toggling around v_wmma), division-free k/v tiling (mi % 4 == 0), branchless
// clamped operand builds (no OOB addresses, no saveexec in the WMMA loop).
// ---------------------------------------------------------------------------

#define Bm 32
#define Nm 24
#define NATOMS (Bm * Nm)        // 768
#define NPAIR (NATOMS * Nm)     // 18432
#define NBASIS 10
#define RMAXF 5.0f
#define BLK 128

typedef __attribute__((ext_vector_type(2))) float v2f;
typedef __attribute__((ext_vector_type(8))) float v8f;

// ------------------------- CG table init (device) --------------------------
__device__ double factd(int n) {
  double r = 1.0;
  for (int i = 2; i <= n; ++i) r *= (double)i;
  return r;
}

__device__ double cg_coeff(int j1, int m1, int j2, int m2, int j3, int m3) {
  if (m1 + m2 != m3) return 0.0;
  double pre = sqrt((2.0 * j3 + 1.0) * factd(j3 + j1 - j2) * factd(j3 - j1 + j2) *
                    factd(j1 + j2 - j3) / factd(j1 + j2 + j3 + 1));
  pre *= sqrt(factd(j3 + m3) * factd(j3 - m3) * factd(j1 - m1) * factd(j1 + m1) *
              factd(j2 - m2) * factd(j2 + m2));
  double s = 0.0;
  for (int k = 0; k <= j1 + j2 + j3; ++k) {
    int a0 = k, a1 = j1 + j2 - j3 - k, a2 = j1 - m1 - k;
    int a3 = j2 + m2 - k, a4 = j3 - j2 + m1 + k, a5 = j3 - j1 - m2 + k;
    if (a0 < 0 || a1 < 0 || a2 < 0 || a3 < 0 || a4 < 0 || a5 < 0) continue;
    double d = factd(a0) * factd(a1) * factd(a2) * factd(a3) * factd(a4) * factd(a5);
    s += ((k & 1) ? -1.0 : 1.0) / d;
  }
  return pre * s;
}

__device__ void u_real(int l, double ur[7][7], double ui[7][7]) {
  for (int r = 0; r < 7; ++r)
    for (int c = 0; c < 7; ++c) { ur[r][c] = 0.0; ui[r][c] = 0.0; }
  double s = 1.0 / sqrt(2.0);
  for (int m = -l; m <= l; ++m) {
    if (m == 0) {
      ur[l][l] = 1.0;
    } else if (m > 0) {
      ur[m + l][m + l] = ((m & 1) ? -1.0 : 1.0) * s;
      ur[m + l][-m + l] = s;
    } else {
      int a = -m;
      ui[m + l][m + l] = s;                              //  i*s
      ui[m + l][a + l] = -(((a & 1) ? -1.0 : 1.0)) * s;  // -i*(-1)^a*s
    }
  }
}

// blockIdx.x encodes (li,lf,lo): slot = ((li*4+lf)*4+lo), 64 blocks.
__global__ void cg_kernel(float* __restrict__ cgout) {
  int bx = blockIdx.x;
  int li = bx >> 4, lf = (bx >> 2) & 3, lo = bx & 3;
  int lfmin = (li > lo) ? (li - lo) : (lo - li);
  int lfmax = (li + lo < 3) ? (li + lo) : 3;
  if (lf < lfmin || lf > lfmax) return;
  int n1 = 2 * li + 1, n2 = 2 * lf + 1, n3 = 2 * lo + 1;
  __shared__ double Tr[343], Ti[343];
  __shared__ double sums[2];
  double u1r[7][7], u1i[7][7], u2r[7][7], u2i[7][7], u3r[7][7], u3i[7][7];
  u_real(li, u1r, u1i);
  u_real(lf, u2r, u2i);
  u_real(lo, u3r, u3i);
  int tot = n1 * n2 * n3;
  for (int e = threadIdx.x; e < tot; e += blockDim.x) {
    int c = e % n3;
    int b = (e / n3) % n2;
    int a = e / (n3 * n2);
    double sr = 0.0, si = 0.0;
    for (int i = 0; i < n1; ++i) {
      for (int j = 0; j < n2; ++j) {
        int m1 = i - li, m2 = j - lf;
        int m3 = m1 + m2;
        if (m3 < -lo || m3 > lo) continue;
        int k = m3 + lo;
        double cgv = cg_coeff(li, m1, lf, m2, lo, m3);
        if (cgv == 0.0) continue;
        double xr = u1r[a][i], xi = u1i[a][i];
        double yr = u2r[b][j], yi = u2i[b][j];
        double pr = xr * yr - xi * yi, pi = xr * yi + xi * yr;
        double zr = u3r[c][k], zi = -u3i[c][k];  // conj(U3)
        sr += (pr * zr - pi * zi) * cgv;
        si += (pr * zi + pi * zr) * cgv;
      }
    }
    Tr[e] = sr;
    Ti[e] = si;
  }
  __syncthreads();
  if (threadIdx.x == 0) {
    double ar = 0.0, ai = 0.0;
    for (int e = 0; e < tot; ++e) { ar += fabs(Tr[e]); ai += fabs(Ti[e]); }
    sums[0] = ar;
    sums[1] = ai;
  }
  __syncthreads();
  bool useR = sums[0] >= sums[1];
  float* dst = cgout + bx * 343;
  for (int e = threadIdx.x; e < tot; e += blockDim.x)
    dst[e] = (float)(useR ? Tr[e] : Ti[e]);
}

// ------------------------- pair geometry kernel ----------------------------
// Y[pair,16] (l offsets 0,1,4,9), rb[pair,10] masked by mask[j].
__global__ void geom_kernel(const float* __restrict__ pos,
                            const float* __restrict__ mask,
                            float* __restrict__ Yg, float* __restrict__ rbg) {
  int p = blockIdx.x * blockDim.x + threadIdx.x;
  if (p >= NPAIR) return;
  int j = p % Nm;
  int a = (p / Nm) % Nm;
  int bb = p / (Nm * Nm);
  const float* pa = pos + (bb * Nm + a) * 3;
  const float* pj = pos + (bb * Nm + j) * 3;
  float dx = pj[0] - pa[0], dy = pj[1] - pa[1], dz = pj[2] - pa[2];
  float r = sqrtf(dx * dx + dy * dy + dz * dz + 1e-12f);
  float invr = 1.0f / (r + 1e-9f);
  float x = dx * invr, y = dy * invr, z = dz * invr;
  float r2 = x * x + y * y + z * z;
  float* Yp = Yg + p * 16;
  Yp[0] = 0.28209479177387814f;
  Yp[1] = 0.4886025119029199f * y;
  Yp[2] = 0.4886025119029199f * z;
  Yp[3] = 0.4886025119029199f * x;
  Yp[4] = 1.0925484305920792f * x * y;
  Yp[5] = 1.0925484305920792f * y * z;
  Yp[6] = 0.31539156525252005f * (3.0f * z * z - r2);
  Yp[7] = 1.0925484305920792f * x * z;
  Yp[8] = 0.5462742152960396f * (x * x - y * y);
  Yp[9] = 0.5900435899266435f * y * (3.0f * x * x - y * y);
  Yp[10] = 2.890611442640554f * x * y * z;
  Yp[11] = 0.4570457994644658f * y * (5.0f * z * z - r2);
  Yp[12] = 0.3731763325901154f * z * (5.0f * z * z - 3.0f * r2);
  Yp[13] = 0.4570457994644658f * x * (5.0f * z * z - r2);
  Yp[14] = 1.445305721320277f * z * (x * x - y * y);
  Yp[15] = 0.5900435899266435f * x * (x * x - 3.0f * y * y);
  float m = mask[bb * Nm + j];
  const float width = RMAXF / (NBASIS - 1);
  float* rp = rbg + p * NBASIS;
  for (int k = 0; k < NBASIS; ++k) {
    float c = (RMAXF * (float)k) / (float)(NBASIS - 1);
    float t = (r - c) / width;
    rp[k] = expf(-0.5f * t * t) * m;
  }
}

// ------------------------------ embedding ----------------------------------
__global__ void embed_kernel(const int* __restrict__ z,
                             const float* __restrict__ embed,
                             float* __restrict__ f0) {
  int e = blockIdx.x * blockDim.x + threadIdx.x;
  if (e >= NATOMS * 32) return;
  int atom = e >> 5, c = e & 31;
  f0[e] = embed[z[atom] * 32 + c];
}

// ------------------------------ layer kernel -------------------------------
// One workgroup per output atom. Phase 1: S[path,k,v,o] in LDS (sum over 24
// neighbors). Phase 2: out[u,o] = W(k,u,v)*S(k,v,o) via v_wmma_f32_16x16x4_f32
// (A = weight tile 16x4, B = S tile 4x16; wave-uniform control flow, branchless
// clamped operand builds -> EXEC all 1s around every WMMA).
// Phase 3: gated nonlinearity + mask, write 128 features/atom.
__global__ void __launch_bounds__(BLK)
layer_kernel(int layer, const float* __restrict__ feat_in,
             const float* __restrict__ radial_w, const float* __restrict__ Yg,
             const float* __restrict__ rbg, const float* __restrict__ cg,
             const float* __restrict__ mask, float* __restrict__ feat_out) {
  __shared__ float S[14600];
  __shared__ float sh_fin[128];
  __shared__ float sh_y[16];
  __shared__ float sh_rb[12];
  __shared__ float sh_out[160];

  const int atom = blockIdx.x;  // b*Nm + a
  const int tid = threadIdx.x;
  const int lane = tid & 31;
  // wave index forced into an SGPR so the unit loop & path filter stay scalar
  const int wave = __builtin_amdgcn_readfirstlane(tid >> 5);

  const int MI_[4] = {24, 12, 8, 4};
  const int FOFF_[4] = {0, 24, 60, 100};
  const int MO_[4] = {48, 12, 8, 4};
  const int OUTOFF_[4] = {0, 48, 84, 124};
  const int YOFF_[4] = {0, 1, 4, 9};
  const int WBASE_[3] = {0, 23040, 68960};

  const int NIN = (layer == 0) ? 1 : 4;
  const int finw = (layer == 0) ? 32 : 128;
  const int S_TOT = (layer == 0) ? 5120 : 14600;
  const int bb = atom / Nm;
  const float inv = 0.23570226039551584f;  // 1/sqrt(18)

  // keep the layer's weights hot (global_prefetch_b8)
  __builtin_prefetch(radial_w + WBASE_[layer], 0, 0);

  for (int e = tid; e < S_TOT; e += BLK) S[e] = 0.0f;
  __syncthreads();

  // ---- Phase 1: accumulate S over neighbors ----
  for (int j = 0; j < Nm; ++j) {
    const float* fj = feat_in + (bb * Nm + j) * finw;
    for (int e = tid; e < finw; e += BLK) sh_fin[e] = fj[e] * inv;
    int pidx = atom * Nm + j;
    if (tid < 16) sh_y[tid] = Yg[pidx * 16 + tid];
    if (tid < NBASIS) sh_rb[tid] = rbg[pidx * NBASIS + tid];
    __syncthreads();

    int soff = 0;
    for (int ii = 0; ii < NIN; ++ii) {
      int li = (layer == 0) ? 0 : ii;
      int mi = (layer == 0) ? 32 : MI_[ii];
      int fo = (layer == 0) ? 0 : FOFF_[ii];
      int n1 = 2 * li + 1;
      for (int lo = 0; lo < 4; ++lo) {
        int no = 2 * lo + 1;
        int lf0 = (li > lo) ? (li - lo) : (lo - li);
        int lf1 = (li + lo < 3) ? (li + lo) : 3;
        for (int lf = lf0; lf <= lf1; ++lf) {
          int n2 = 2 * lf + 1;
          const float* Q = cg + ((li * 4 + lf) * 4 + lo) * 343;
          const float* yb = sh_y + YOFF_[lf];
          int ne = mi * no;
          for (int e = tid; e < ne; e += BLK) {
            int v = e / no, o = e - v * no;
            const float* fv = sh_fin + fo + v * n1;
            float t = 0.0f;
            for (int i = 0; i < n1; ++i) {
              float fi = fv[i];
              const float* Qi = Q + (i * n2) * no + o;
              for (int f = 0; f < n2; ++f) t += fi * yb[f] * Qi[f * no];
            }
            float* Sp = S + soff + v * no + o;
            for (int k = 0; k < NBASIS; ++k) Sp[k * mi * no] += sh_rb[k] * t;
          }
          soff += NBASIS * mi * no;
        }
      }
    }
    __syncthreads();
  }

  // ---- Phase 2: W contraction via f32 WMMA ----
  for (int e = tid; e < 160; e += BLK) sh_out[e] = 0.0f;
  __syncthreads();

  const int row = lane & 15;
  const int hi = lane >> 4;

  // 6 units: (lo=0, tiles 0..2), (lo=1), (lo=2), (lo=3); 4 waves round-robin.
  for (int unit = wave; unit < 6; unit += 4) {
    const int ulo = (unit < 3) ? 0 : (unit - 2);
    const int tile = (unit < 3) ? unit : 0;
    const int mo = MO_[ulo];
    const int no = 2 * ulo + 1;

    // lane-level clamps (branchless; no saveexec, no OOB addresses)
    const int u = tile * 16 + row;
    const int uc = (u < mo) ? u : 0;         // clamped row index into W
    const int oc = (row < no) ? row : 0;     // clamped col index into S
    const float bmask = (row < no) ? 1.0f : 0.0f;

    v8f acc;
    for (int r = 0; r < 8; ++r) acc[r] = 0.0f;

    int woff = WBASE_[layer];
    int soff = 0;
    for (int ii = 0; ii < NIN; ++ii) {
      int li = (layer == 0) ? 0 : ii;
      int mi = (layer == 0) ? 32 : MI_[ii];
      for (int lo = 0; lo < 4; ++lo) {
        int no2 = 2 * lo + 1;
        int mo2 = MO_[lo];
        int lf0 = (li > lo) ? (li - lo) : (lo - li);
        int lf1 = (li + lo < 3) ? (li + lo) : 3;
        for (int lf = lf0; lf <= lf1; ++lf) {
          if (lo == ulo) {  // scalar branch (unit/ulo are wave-uniform SGPRs)
            const float amask = ((u < mo) ? 1.0f : 0.0f) * rsqrtf((float)(NBASIS * mi));
            const int nvc = mi >> 2;  // mi in {32,24,12,8,4}: always /4
            for (int k = 0; k < NBASIS; ++k) {
              const float* Wk = radial_w + woff + (k * mo + uc) * mi;
              const float* Sk = S + soff + k * mi * no;
              for (int vc = 0; vc < nvc; ++vc) {
                const int v0 = vc * 4 + hi * 2;  // lanes 0-15: K=0,1; 16-31: K=2,3
                v2f a, b;
                a.x = Wk[v0] * amask;
                a.y = Wk[v0 + 1] * amask;
                b.x = Sk[v0 * no + oc] * bmask;
                b.y = Sk[(v0 + 1) * no + oc] * bmask;
                acc = __builtin_amdgcn_wmma_f32_16x16x4_f32(
                    false, a, false, b, (short)0, acc, false, false);
              }
            }
          }
          soff += NBASIS * mi * no2;
          woff += NBASIS * mo2 * mi;
        }
      }
    }
    // D layout: lane L, VGPR r -> M = r + 8*(L>=16), N = L%16
    for (int r = 0; r < 8; ++r) {
      int uu = tile * 16 + r + hi * 8;
      int oo = row;
      if (uu < mo && oo < no) sh_out[OUTOFF_[ulo] + uu * no + oo] = acc[r];
    }
  }
  __syncthreads();

  // ---- Phase 3: gated nonlinearity + mask, 128 outputs/atom ----
  {
    float mval = mask[atom];
    int c = tid;
    float res;
    if (c < 24) {
      float s = sh_out[c];
      res = (s > 0.0f) ? s : 0.0f;
    } else if (c < 60) {
      int idx = c - 24;
      int u = idx / 3;
      float g = 1.0f / (1.0f + expf(-sh_out[24 + u]));
      res = sh_out[48 + idx] * g;
    } else if (c < 100) {
      int idx = c - 60;
      int u = idx / 5;
      float g = 1.0f / (1.0f + expf(-sh_out[36 + u]));
      res = sh_out[84 + idx] * g;
    } else {
      int idx = c - 100;
      int u = idx / 7;
      float g = 1.0f / (1.0f + expf(-sh_out[44 + u]));
      res = sh_out[124 + idx] * g;
    }
    feat_out[atom * 128 + c] = res * mval;
  }
}

// ------------------------------- launcher ----------------------------------
extern "C" void kernel_launch(void* const* d_in, const int* in_sizes, int n_in,
                              void* d_out, int out_size, void* d_ws,
                              size_t ws_size, hipStream_t stream) {
  const int* z = (const int*)d_in[0];
  const float* pos = (const float*)d_in[1];
  const float* mask = (const float*)d_in[2];
  const float* embed = (const float*)d_in[3];
  const float* rw = (const float*)d_in[4];

  float* ws = (float*)d_ws;
  float* cg = ws;                       // 64*343      = 21952
  float* Yg = cg + 21952;               // 18432*16    = 294912
  float* rbg = Yg + 294912;             // 18432*10    = 184320
  float* f0 = rbg + 184320;             // 768*32      = 24576
  float* fA = f0 + 24576;               // 768*128     = 98304
  float* fB = fA + 98304;               // 768*128     = 98304
  float* out = (float*)d_out;           // 768*128

  cg_kernel<<<64, 64, 0, stream>>>(cg);
  geom_kernel<<<NPAIR / BLK, BLK, 0, stream>>>(pos, mask, Yg, rbg);
  embed_kernel<<<(NATOMS * 32) / BLK, BLK, 0, stream>>>(z, embed, f0);
  layer_kernel<<<NATOMS, BLK, 0, stream>>>(0, f0, rw, Yg, rbg, cg, mask, fA);
  layer_kernel<<<NATOMS, BLK, 0, stream>>>(1, fA, rw, Yg, rbg, cg, mask, fB);
  layer_kernel<<<NATOMS, BLK, 0, stream>>>(2, fB, rw, Yg, rbg, cg, mask, out);
}